// QFCModel_2104533975794
// MI455X (gfx1250) — compile-verified
//
#include <hip/hip_runtime.h>
#include <stdint.h>

// ===========================================================================
// Compile-time reproduction of np.random.default_rng(42) RandomLayer structure
// (SeedSequence pool mix -> PCG64 XSL-RR 128/64 with buffered next32 ->
//  Generator.integers (Lemire) / Generator.choice (Floyd + interval-shuffle))
// ===========================================================================
typedef unsigned int       u32;
typedef unsigned long long u64;
typedef __uint128_t        u128;

struct RngState { u128 state; u128 inc; bool has32; u32 buf32; };

constexpr u64 rotr64c(u64 v, unsigned r) { r &= 63u; return (v >> r) | (v << ((64u - r) & 63u)); }

constexpr void pcg_step(RngState& g) {
  const u128 mult = ((u128)2549297995355413924ULL << 64) | (u128)4865540595714422341ULL;
  g.state = g.state * mult + g.inc;
}
constexpr u64 next64(RngState& g) {
  pcg_step(g);
  const u64 hi = (u64)(g.state >> 64), lo = (u64)g.state;
  return rotr64c(hi ^ lo, (unsigned)(g.state >> 122));
}
constexpr u32 next32(RngState& g) {
  if (g.has32) { g.has32 = false; return g.buf32; }
  const u64 n = next64(g);
  g.has32 = true; g.buf32 = (u32)(n >> 32);
  return (u32)n;
}
// Lemire bounded draw on [0, rmax] (numpy Generator.integers path, 32-bit range)
constexpr u32 lemire32(RngState& g, u32 rmax) {
  const u32 re = rmax + 1u;
  u64 m = (u64)next32(g) * (u64)re;
  u32 left = (u32)m;
  if (left < re) {
    const u32 th = (0xFFFFFFFFu - rmax) % re;
    while (left < th) { m = (u64)next32(g) * (u64)re; left = (u32)m; }
  }
  return (u32)(m >> 32);
}
// numpy random_interval (masked rejection) for max <= 2^32-1 (used by shuffle)
constexpr u32 masked_interval(RngState& g, u32 mx) {
  if (!mx) return 0;
  u32 mask = mx; mask |= mask >> 1; mask |= mask >> 2; mask |= mask >> 4; mask |= mask >> 8; mask |= mask >> 16;
  u32 v = next32(g) & mask;
  while (v > mx) v = next32(g) & mask;
  return v;
}

constexpr RngState make_rng42() {
  // SeedSequence(42): pool_size=4, entropy=[42]
  u32 pool[4] = {0, 0, 0, 0};
  u32 hc = 0x43b0d7e5u;                      // INIT_A
  for (int i = 0; i < 4; i++) {
    u32 v = (i == 0) ? 42u : 0u;
    v ^= hc; hc *= 0x931e8875u; v *= hc; v ^= v >> 16;   // hashmix
    pool[i] = v;
  }
  for (int s = 0; s < 4; s++)
    for (int d = 0; d < 4; d++)
      if (s != d) {
        u32 v = pool[s];
        v ^= hc; hc *= 0x931e8875u; v *= hc; v ^= v >> 16;              // hashmix
        u32 r = pool[d] * 0xca01f9ddu - v * 0x4973f715u; r ^= r >> 16;  // mix
        pool[d] = r;
      }
  // generate_state(4, uint64) = 8 uint32 words, little-endian pairs
  u32 hb = 0x8b51f9ddu;                      // INIT_B
  u64 o[4] = {0, 0, 0, 0};
  for (int k = 0; k < 8; k++) {
    u32 v = pool[k & 3];
    v ^= hb; hb *= 0x58f38dedu; v *= hb; v ^= v >> 16;
    o[k >> 1] |= (u64)v << ((k & 1) * 32);
  }
  RngState g{};
  const u128 initstate = ((u128)o[0] << 64) | o[1];   // seed[0] is HIGH word
  const u128 initseq   = ((u128)o[2] << 64) | o[3];
  g.state = 0; g.inc = (initseq << 1) | (u128)1; g.has32 = false; g.buf32 = 0;
  pcg_step(g); g.state += initstate; pcg_step(g);
  return g;
}

struct OpsTable { int t[50]; int w0[50]; int w1[50]; int pi[50]; int nparams; };

constexpr OpsTable make_ops() {
  OpsTable T{};
  RngState g = make_rng42();
  int p = 0;
  for (int o = 0; o < 50; o++) {
    const u32 t = lemire32(g, 3u);
    if (t == 3u) {
      // rng.choice(4, 2, replace=False): Floyd's algorithm, hash set of 4, then shuffle
      long long hs[4] = {-1, -1, -1, -1};
      int idx[2] = {0, 0};
      for (int j = 2; j <= 3; j++) {
        const u64 val = (u64)lemire32(g, (u32)j);
        u32 loc = (u32)val & 3u;
        while (hs[loc] != -1 && (u64)hs[loc] != val) loc = (loc + 1u) & 3u;
        if (hs[loc] == -1) { hs[loc] = (long long)val; idx[j - 2] = (int)val; }
        else {
          loc = (u32)j & 3u;
          while (hs[loc] != -1) loc = (loc + 1u) & 3u;
          hs[loc] = j; idx[j - 2] = j;
        }
      }
      const u32 jj = masked_interval(g, 1u);   // Fisher-Yates step i=1
      if (jj == 0u) { const int tmp = idx[1]; idx[1] = idx[0]; idx[0] = tmp; }
      T.t[o] = 3; T.w0[o] = idx[0]; T.w1[o] = idx[1]; T.pi[o] = -1;
    } else {
      const u32 w = lemire32(g, 3u);
      T.t[o] = (int)t; T.w0[o] = (int)w; T.w1[o] = -1; T.pi[o] = p++;
    }
  }
  T.nparams = p;
  return T;
}

__constant__ OpsTable c_ops = make_ops();

// ===========================================================================
// Kernel 1: build the fixed 16x16 complex unitary W = H(wire1) * prod(RL ops).
// The real 32x32 matrix A with [re(out)|im(out)] = [re(s)|im(s)] @ A is
//   A[i   , j   ] =  Wr[j][i]     A[16+i, j   ] = -Wi[j][i]
//   A[i   ,16+j ] =  Wi[j][i]     A[16+i,16+j ] =  Wr[j][i]
// Emitted PRE-SWIZZLED into V_WMMA_F32_16X16X4_F32 B-operand order:
//   Bsw[(k*2+n)*32 + lane] = { A[4k + 2*(lane>=16) + 0][n*16 + (lane&15)],
//                              A[4k + 2*(lane>=16) + 1][n*16 + (lane&15)] }
// Wire w lives at flat-index bit (3 - w)  =>  pair stride = 8 >> w.
// ===========================================================================
__global__ __launch_bounds__(256) void build_unitary_kernel(
    const float* __restrict__ rl, float2* __restrict__ Bsw) {
  __shared__ float shr[16][17];   // [column j][amplitude k]
  __shared__ float shi[16][17];
  const int tid = threadIdx.x;
  if (tid < 16) {
    for (int k = 0; k < 16; k++) { shr[tid][k] = (k == tid) ? 1.0f : 0.0f; shi[tid][k] = 0.0f; }
    for (int o = 0; o < 50; o++) {
      const int t = c_ops.t[o];
      if (t == 3) {                       // CNOT(control=w0, target=w1)
        const int sc = 8 >> c_ops.w0[o], st = 8 >> c_ops.w1[o];
        for (int k = 0; k < 16; k++) {
          if ((k & sc) && !(k & st)) {
            const int k2 = k | st;
            float tr = shr[tid][k]; shr[tid][k] = shr[tid][k2]; shr[tid][k2] = tr;
            float ti = shi[tid][k]; shi[tid][k] = shi[tid][k2]; shi[tid][k2] = ti;
          }
        }
      } else {                            // RX / RY / RZ (theta = rl_params[pi])
        float sn, cs;
        sincosf(0.5f * rl[c_ops.pi[o]], &sn, &cs);
        const int st = 8 >> c_ops.w0[o];
        for (int k = 0; k < 16; k++) {
          if (!(k & st)) {
            const int k2 = k | st;
            const float ar = shr[tid][k], ai = shi[tid][k];
            const float br = shr[tid][k2], bi = shi[tid][k2];
            float nr0, ni0, nr1, ni1;
            if (t == 0) {        // RX: [[c,-is],[-is,c]]
              nr0 =  cs * ar + sn * bi;  ni0 =  cs * ai - sn * br;
              nr1 =  sn * ai + cs * br;  ni1 = -sn * ar + cs * bi;
            } else if (t == 1) { // RY: [[c,-s],[s,c]]
              nr0 = cs * ar - sn * br;   ni0 = cs * ai - sn * bi;
              nr1 = sn * ar + cs * br;   ni1 = sn * ai + cs * bi;
            } else {             // RZ: diag(c-is, c+is)
              nr0 = cs * ar + sn * ai;   ni0 = cs * ai - sn * ar;
              nr1 = cs * br - sn * bi;   ni1 = cs * bi + sn * br;
            }
            shr[tid][k] = nr0; shi[tid][k] = ni0;
            shr[tid][k2] = nr1; shi[tid][k2] = ni1;
          }
        }
      }
    }
    // Final H on wire 1 (stride 4)
    const float rh = 0.70710678118654752f;
    for (int k = 0; k < 16; k++) {
      if (!(k & 4)) {
        const int k2 = k | 4;
        const float ar = shr[tid][k], ai = shi[tid][k];
        const float br = shr[tid][k2], bi = shi[tid][k2];
        shr[tid][k] = rh * (ar + br);  shi[tid][k] = rh * (ai + bi);
        shr[tid][k2] = rh * (ar - br); shi[tid][k2] = rh * (ai - bi);
      }
    }
  }
  __syncthreads();

  auto Aval = [&](int i, int jc) -> float {
    if (jc < 16) return (i < 16) ? shr[i][jc] : -shi[i - 16][jc];
    const int j = jc - 16;
    return (i < 16) ? shi[i][j] : shr[i - 16][j];
  };
  for (int p = tid; p < 512; p += blockDim.x) {
    const int k = p >> 6;            // K-step 0..7
    const int n = (p >> 5) & 1;      // output block 0..1
    const int l = p & 31;            // consuming lane
    const int row0 = 4 * k + ((l >> 4) ? 2 : 0);
    const int col  = n * 16 + (l & 15);
    float2 v; v.x = Aval(row0, col); v.y = Aval(row0 + 1, col);
    Bsw[p] = v;
  }
}

// ===========================================================================
// Kernel 2: per-wave 16-sample tile. pool -> encode (product state) ->
// [re|im] (16x32) x A (32x32) via 16 chained V_WMMA_F32_16X16X4_F32 ->
// probs -> <Z> -> logits -> log_softmax.
// ===========================================================================
typedef float v2f __attribute__((ext_vector_type(2)));
typedef float v8f __attribute__((ext_vector_type(8)));

struct cpx { float x, y; };
__device__ inline cpx cmul(cpx a, cpx b) { return cpx{a.x * b.x - a.y * b.y, a.x * b.y + a.y * b.x}; }

__global__ __launch_bounds__(256) void qfc_forward_kernel(
    const float* __restrict__ x, const float2* __restrict__ Bsw,
    float* __restrict__ out, int B) {
  __shared__ float2 sB[512];
  for (int e = threadIdx.x; e < 512; e += 256) sB[e] = Bsw[e];
  __syncthreads();

  const int lane = threadIdx.x & 31;
  const int wave = threadIdx.x >> 5;
  const int base = (blockIdx.x * 8 + wave) * 16;   // wave-uniform
  if (base >= B) return;
  const int hi = lane >> 4;       // which K-half of the WMMA operands this lane feeds
  const int m  = lane & 15;       // sample row within tile / output column
  int samp = base + m; if (samp >= B) samp = B - 1;

  // ---- 6x6 average pool over leading 24x24 (rows 16B-aligned: 28 floats) ----
  const float4* __restrict__ xr = reinterpret_cast<const float4*>(x + (size_t)samp * 784);
  float pooled[4][4];
#pragma unroll
  for (int rb = 0; rb < 4; ++rb) {
    float a0 = 0.f, a1 = 0.f, a2 = 0.f, a3 = 0.f;
#pragma unroll
    for (int rr = 0; rr < 6; ++rr) {
      const float4* row = xr + (rb * 6 + rr) * 7;
      const float4 q0 = row[0], q1 = row[1], q2 = row[2], q3 = row[3], q4 = row[4], q5 = row[5];
      a0 += q0.x + q0.y + q0.z + q0.w + q1.x + q1.y;
      a1 += q1.z + q1.w + q2.x + q2.y + q2.z + q2.w;
      a2 += q3.x + q3.y + q3.z + q3.w + q4.x + q4.y;
      a3 += q4.z + q4.w + q5.x + q5.y + q5.z + q5.w;
    }
    const float inv = 1.0f / 36.0f;
    pooled[rb][0] = a0 * inv; pooled[rb][1] = a1 * inv;
    pooled[rb][2] = a2 * inv; pooled[rb][3] = a3 * inv;
  }

  // ---- encoder: per wire  v_w = RX(p3) RZ(p2) RY(p1) RX(p0) |0> ----
  cpx vw[4][2];
#pragma unroll
  for (int w = 0; w < 4; ++w) {
    float s0, c0, s1, c1, s2, c2, s3, c3;
    sincosf(0.5f * pooled[0][w], &s0, &c0);
    sincosf(0.5f * pooled[1][w], &s1, &c1);
    sincosf(0.5f * pooled[2][w], &s2, &c2);
    sincosf(0.5f * pooled[3][w], &s3, &c3);
    const cpx e0{c0, 0.f}, e1{0.f, -s0};                        // RX(p0)|0>
    const cpx f0{c1 * e0.x - s1 * e1.x, c1 * e0.y - s1 * e1.y}; // RY
    const cpx f1{s1 * e0.x + c1 * e1.x, s1 * e0.y + c1 * e1.y};
    const cpx g0 = cmul(f0, cpx{c2, -s2});                      // RZ
    const cpx g1 = cmul(f1, cpx{c2,  s2});
    vw[w][0] = cpx{c3 * g0.x + s3 * g1.y,  c3 * g0.y - s3 * g1.x};   // RX(p3)
    vw[w][1] = cpx{s3 * g0.y + c3 * g1.x, -s3 * g0.x + c3 * g1.y};
  }

  // ---- tensor product, restricted to this lane's K-half ----
  // amplitude index k = 4h + l (h over wires 0,1; l over wires 2,3).
  // Lanes 0-15 feed K columns 4k+{0,1} -> need l in {0,1};
  // lanes 16-31 feed K columns 4k+{2,3} -> need l in {2,3}.
  cpx v01[4], v23[4];
#pragma unroll
  for (int a = 0; a < 2; a++)
#pragma unroll
    for (int b = 0; b < 2; b++) {
      v01[a * 2 + b] = cmul(vw[0][a], vw[1][b]);
      v23[a * 2 + b] = cmul(vw[2][a], vw[3][b]);
    }
  cpx t0, t1;   // two cndmask pairs instead of a dynamic-index select tree
  t0.x = hi ? v23[2].x : v23[0].x;  t0.y = hi ? v23[2].y : v23[0].y;
  t1.x = hi ? v23[3].x : v23[1].x;  t1.y = hi ? v23[3].y : v23[1].y;
  cpx amp0[4], amp1[4];             // amplitudes (4h + ofs), (4h + ofs + 1)
#pragma unroll
  for (int h = 0; h < 4; ++h) { amp0[h] = cmul(v01[h], t0); amp1[h] = cmul(v01[h], t1); }

  // ---- O(16x32) = S(16x32) x A(32x32): 8 K-steps x 2 output blocks ----
  // K-steps 0..3 carry re(amp), 4..7 carry im(amp).
  v8f acc0 = {0, 0, 0, 0, 0, 0, 0, 0};   // out_re columns 0..15
  v8f acc1 = {0, 0, 0, 0, 0, 0, 0, 0};   // out_im columns 0..15
#pragma unroll
  for (int k = 0; k < 8; k++) {
    v2f a;
    if (k < 4) { a.x = amp0[k].x;     a.y = amp1[k].x; }
    else       { a.x = amp0[k - 4].y; a.y = amp1[k - 4].y; }
    const float2 bb0 = sB[(k * 2 + 0) * 32 + lane];
    const float2 bb1 = sB[(k * 2 + 1) * 32 + lane];
    v2f b0; b0.x = bb0.x; b0.y = bb0.y;
    v2f b1; b1.x = bb1.x; b1.y = bb1.y;
    acc0 = __builtin_amdgcn_wmma_f32_16x16x4_f32(false, a, false, b0, (short)0, acc0, false, false);
    acc1 = __builtin_amdgcn_wmma_f32_16x16x4_f32(false, a, false, b1, (short)0, acc1, false, false);
  }

  // ---- epilogue: D VGPR v holds row (v + 8*hi), column k = m ----
  const float c0s = 2.f - 2.f * ((m >> 3) & 1) - 2.f * ((m >> 2) & 1);  // wires 0+1
  const float c1s = 2.f - 2.f * ((m >> 1) & 1) - 2.f * (m & 1);         // wires 2+3
  float l0[8], l1[8];
#pragma unroll
  for (int v = 0; v < 8; ++v) {
    const float p = acc0[v] * acc0[v] + acc1[v] * acc1[v];
    l0[v] = p * c0s; l1[v] = p * c1s;
  }
  // wave32 butterflies over the 16 amplitude lanes (masks < 16 stay in half)
#pragma unroll
  for (int mask = 1; mask < 16; mask <<= 1) {
#pragma unroll
    for (int v = 0; v < 8; ++v) {
      l0[v] += __shfl_xor(l0[v], mask, 32);
      l1[v] += __shfl_xor(l1[v], mask, 32);
    }
  }
  float r0 = l0[0], r1 = l1[0];
#pragma unroll
  for (int v = 1; v < 8; ++v) { if (m == v) { r0 = l0[v]; r1 = l1[v]; } }
  if (m < 8) {
    const int srow = base + hi * 8 + m;
    if (srow < B) {
      const float mx = fmaxf(r0, r1);
      const float lse = mx + logf(expf(r0 - mx) + expf(r1 - mx));
      out[srow * 2 + 0] = r0 - lse;
      out[srow * 2 + 1] = r1 - lse;
    }
  }
}

// ===========================================================================
extern "C" void kernel_launch(void* const* d_in, const int* in_sizes, int n_in,
                              void* d_out, int out_size, void* d_ws, size_t ws_size,
                              hipStream_t stream) {
  (void)n_in; (void)out_size; (void)ws_size;
  const float* x  = (const float*)d_in[0];
  const float* rl = (const float*)d_in[1];
  float* out = (float*)d_out;
  float2* Bsw = (float2*)d_ws;               // 512 float2 of scratch (4 KB)
  const int B = in_sizes[0] / 784;

  build_unitary_kernel<<<1, 256, 0, stream>>>(rl, Bsw);
  const int blocks = (B + 127) / 128;        // 8 waves/block, 16 samples/wave
  qfc_forward_kernel<<<blocks, 256, 0, stream>>>(x, Bsw, out, B);
}